// MoELayer_62749472194930
// MI455X (gfx1250) — compile-verified
//
#include <hip/hip_runtime.h>
#include <math.h>

#define D_MODEL     1024
#define D_FF        4096
#define NUM_EXPERTS 24
#define TOKENS      2048
#define MAXT        2048   // per-expert token list capacity (worst case)

typedef __attribute__((ext_vector_type(2))) float v2f;
typedef __attribute__((ext_vector_type(8))) float v8f;

// ---------------------------------------------------------------------------
// Kernel 1: out = x (residual base, vectorized copy), zero per-expert counters
// ---------------------------------------------------------------------------
__global__ __launch_bounds__(256) void moe_init(const float4* __restrict__ x4,
                                                float4* __restrict__ out4,
                                                int* __restrict__ cnt) {
    int i = blockIdx.x * blockDim.x + threadIdx.x;
    if (i < TOKENS * D_MODEL / 4) out4[i] = x4[i];
    if (i < NUM_EXPERTS) cnt[i] = 0;
}

// ---------------------------------------------------------------------------
// Kernel 2: gating. One wave (32 threads) per token. Lanes 0..23 compute the
// 24 gate scores; lane 0 does top-2 + softmax and appends (token, gate) to the
// two winning experts' lists.
// ---------------------------------------------------------------------------
__global__ __launch_bounds__(32) void moe_gate(const float* __restrict__ x,
                                               const float* __restrict__ gw,
                                               const float* __restrict__ gb,
                                               int* __restrict__ cnt,
                                               int* __restrict__ tokIdx,
                                               float* __restrict__ tokGate) {
    __shared__ float sc[NUM_EXPERTS];
    const int t = blockIdx.x;
    const int e = threadIdx.x;
    if (e < NUM_EXPERTS) {
        float s = gb[e];
        const float* xr = x + t * D_MODEL;
        for (int d = 0; d < D_MODEL; ++d)
            s = fmaf(xr[d], gw[d * NUM_EXPERTS + e], s);
        sc[e] = s;
    }
    __syncthreads();
    if (threadIdx.x == 0) {
        int i0 = 0; float v0 = sc[0];
        for (int i = 1; i < NUM_EXPERTS; ++i)
            if (sc[i] > v0) { v0 = sc[i]; i0 = i; }
        int i1 = (i0 == 0) ? 1 : 0; float v1 = sc[i1];
        for (int i = 0; i < NUM_EXPERTS; ++i)
            if (i != i0 && sc[i] > v1) { v1 = sc[i]; i1 = i; }
        // softmax over the two winners (v0 >= v1, numerically stable)
        float g0 = 1.0f / (1.0f + __expf(v1 - v0));
        float g1 = 1.0f - g0;
        int s0 = atomicAdd(&cnt[i0], 1);
        tokIdx[i0 * MAXT + s0]  = t;
        tokGate[i0 * MAXT + s0] = g0;
        int s1 = atomicAdd(&cnt[i1], 1);
        tokIdx[i1 * MAXT + s1]  = t;
        tokGate[i1 * MAXT + s1] = g1;
    }
}

// ---------------------------------------------------------------------------
// Kernel 3: fused expert FFN using V_WMMA_F32_16X16X4_F32.
// grid = (128 token-tiles, 24 experts); 256 threads = 8 waves.
// Block processes 16 gathered tokens of expert e:
//   H(16x4096) = gelu(X*W1 + b1) computed in 128-col chunks (H chunk in LDS),
//   Y(16x1024) += Hchunk * W2chunk accumulated in registers (8 x v8f / wave),
//   out[token] += gate * (Y + b2) via f32 atomics.
// ---------------------------------------------------------------------------
__global__ __launch_bounds__(256) void moe_ffn(const float* __restrict__ x,
                                               const float* __restrict__ w1,
                                               const float* __restrict__ b1,
                                               const float* __restrict__ w2,
                                               const float* __restrict__ b2,
                                               const int* __restrict__ cnt,
                                               const int* __restrict__ tokIdx,
                                               const float* __restrict__ tokGate,
                                               float* __restrict__ out) {
    __shared__ float Xs[16 * D_MODEL];   // 64 KB gathered activations
    __shared__ float Hs[16 * 128];       // 8 KB  H chunk
    __shared__ int   sTok[16];
    __shared__ float sGate[16];

    const int e    = blockIdx.y;
    const int tile = blockIdx.x;
    const int cntE = cnt[e];
    if (tile * 16 >= cntE) return;       // uniform per block: EXEC stays full

    const int tid  = threadIdx.x;
    const int w    = tid >> 5;           // wave id 0..7
    const int lane = tid & 31;
    const int half = lane >> 4;          // 0: lanes 0-15, 1: lanes 16-31
    const int l16  = lane & 15;

    if (tid < 16) {
        int r = tile * 16 + tid;
        if (r < cntE) {
            sTok[tid]  = tokIdx[e * MAXT + r];
            sGate[tid] = tokGate[e * MAXT + r];
        } else {
            sTok[tid]  = 0;              // padded row: real data, zero gate
            sGate[tid] = 0.0f;
        }
    }
    __syncthreads();

    // stage X tile with 128-bit loads/stores: 16 rows x 256 float4
    {
        float4* Xs4 = (float4*)Xs;
        for (int i = tid; i < 16 * (D_MODEL / 4); i += 256) {
            int row = i >> 8;            // 256 float4 per row
            int col = i & 255;
            Xs4[i] = ((const float4*)(x + (size_t)sTok[row] * D_MODEL))[col];
        }
    }
    __syncthreads();

    const float* W1 = w1 + (size_t)e * D_MODEL * D_FF;
    const float* W2 = w2 + (size_t)e * D_FF * D_MODEL;

    v8f yacc[8];
    #pragma unroll
    for (int nt = 0; nt < 8; ++nt)
        #pragma unroll
        for (int r = 0; r < 8; ++r) yacc[nt][r] = 0.0f;

    #pragma unroll 1
    for (int chunk = 0; chunk < D_FF / 128; ++chunk) {
        const int fbase = chunk * 128;
        const int fcol  = fbase + w * 16 + l16;   // wave w owns 16 H columns

        // ---- GEMM1: H tile (16 tokens x 16 f-cols), K = 1024 ----
        v8f hacc;
        #pragma unroll
        for (int r = 0; r < 8; ++r) hacc[r] = 0.0f;

        #pragma unroll 4
        for (int k = 0; k < D_MODEL; k += 4) {
            const int ka = k + 2 * half;          // A/B K index per ISA layout
            v2f a;
            a.x = Xs[l16 * D_MODEL + ka];
            a.y = Xs[l16 * D_MODEL + ka + 1];
            const float* pb = &W1[(size_t)ka * D_FF + fcol];
            v2f b;
            b.x = pb[0];
            b.y = pb[D_FF];
            // speculative prefetch of next F-chunk's B cachelines
            // (same rows, +128 columns) -> global_prefetch_b8
            __builtin_prefetch(pb + 128, 0, 1);
            hacc = __builtin_amdgcn_wmma_f32_16x16x4_f32(
                false, a, false, b, (short)0, hacc, false, false);
        }

        // bias + exact gelu, store H chunk to LDS
        const float bias1 = b1[e * D_FF + fcol];
        #pragma unroll
        for (int r = 0; r < 8; ++r) {
            float v = hacc[r] + bias1;            // element (M = r+8*half, N = l16)
            v = 0.5f * v * (1.0f + erff(v * 0.70710678118654752f));
            Hs[(r + 8 * half) * 128 + (w * 16 + l16)] = v;
        }
        __syncthreads();

        // ---- GEMM2 partial: Y[:, w*128 .. +127] += Hchunk(16x128) @ W2chunk ----
        #pragma unroll 2
        for (int kk = 0; kk < 128; kk += 4) {
            const int ka = kk + 2 * half;
            v2f a;
            a.x = Hs[l16 * 128 + ka];
            a.y = Hs[l16 * 128 + ka + 1];
            const int kg = fbase + ka;
            const float* pb = W2 + (size_t)kg * D_MODEL + w * 128 + l16;
            #pragma unroll
            for (int nt = 0; nt < 8; ++nt) {
                v2f b;
                b.x = pb[nt * 16];
                b.y = pb[nt * 16 + D_MODEL];
                yacc[nt] = __builtin_amdgcn_wmma_f32_16x16x4_f32(
                    false, a, false, b, (short)0, yacc[nt], false, false);
            }
        }
        __syncthreads();
    }

    // ---- epilogue: out[token] += gate * (Y + b2) ----
    #pragma unroll
    for (int nt = 0; nt < 8; ++nt) {
        const int col   = w * 128 + nt * 16 + l16;
        const float bias2 = b2[e * D_MODEL + col];
        #pragma unroll
        for (int r = 0; r < 8; ++r) {
            const int row = r + 8 * half;
            const float g = sGate[row];
            if (g != 0.0f) {
                atomicAdd(&out[(size_t)sTok[row] * D_MODEL + col],
                          g * (yacc[nt][r] + bias2));
            }
        }
    }
}

// ---------------------------------------------------------------------------
extern "C" void kernel_launch(void* const* d_in, const int* in_sizes, int n_in,
                              void* d_out, int out_size, void* d_ws, size_t ws_size,
                              hipStream_t stream) {
    const float* x  = (const float*)d_in[0];
    const float* gw = (const float*)d_in[1];
    const float* gb = (const float*)d_in[2];
    const float* w1 = (const float*)d_in[3];
    const float* b1 = (const float*)d_in[4];
    const float* w2 = (const float*)d_in[5];
    const float* b2 = (const float*)d_in[6];
    float* out = (float*)d_out;

    // workspace layout: [cnt: 24 ints, padded to 256B][tokIdx: 24*2048 ints]
    //                   [tokGate: 24*2048 floats]  (~394 KB total)
    char* ws = (char*)d_ws;
    int*   cnt     = (int*)ws;
    int*   tokIdx  = (int*)(ws + 256);
    float* tokGate = (float*)(ws + 256 + (size_t)NUM_EXPERTS * MAXT * sizeof(int));

    moe_init<<<(TOKENS * D_MODEL / 4 + 255) / 256, 256, 0, stream>>>(
        (const float4*)x, (float4*)out, cnt);
    moe_gate<<<TOKENS, 32, 0, stream>>>(x, gw, gb, cnt, tokIdx, tokGate);
    dim3 grid(TOKENS / 16, NUM_EXPERTS);
    moe_ffn<<<grid, 256, 0, stream>>>(x, w1, b1, w2, b2, cnt, tokIdx, tokGate, out);
}